// myLayer_82952998355819
// MI455X (gfx1250) — compile-verified
//
#include <hip/hip_runtime.h>
#include <math.h>

typedef float v2f __attribute__((ext_vector_type(2)));
typedef float v8f __attribute__((ext_vector_type(8)));

#define NEG_SLOPE 0.2f
#define D 128

// ---- ordered-uint encoding for float atomic max (monotonic over all finite floats) ----
__device__ __forceinline__ unsigned f2ord(float f) {
    unsigned u = __float_as_uint(f);
    return (u & 0x80000000u) ? ~u : (u | 0x80000000u);
}
__device__ __forceinline__ float ord2f(unsigned u) {
    return (u & 0x80000000u) ? __uint_as_float(u & 0x7FFFFFFFu) : __uint_as_float(~u);
}
// f2ord(-inf) == 0x007FFFFF  -> init value for segment max

// ---------------------------------------------------------------------------
// Kernel 0: initialize accumulators
// ---------------------------------------------------------------------------
__global__ void k_init(unsigned* __restrict__ segmax, float* __restrict__ denom,
                       float* __restrict__ deg, float* __restrict__ agg,
                       int N, int ND) {
    int i = blockIdx.x * blockDim.x + threadIdx.x;
    if (i < N) {
        segmax[i] = 0x007FFFFFu;   // encoded -inf
        denom[i]  = 0.0f;
        deg[i]    = 0.0f;
    }
    if (i < ND) agg[i] = 0.0f;
}

// ---------------------------------------------------------------------------
// Kernel 1: dual GEMM via fp32 WMMA.  blockIdx.x = row tile (16 rows),
// blockIdx.y = which weight matrix (0: W -> hout w/ NaN scrub, 1: res_w -> res + bias).
// 8 waves per block, wave w owns output column tile [16w, 16w+16).
// K=128 -> 32 x V_WMMA_F32_16X16X4_F32 per wave.
// ---------------------------------------------------------------------------
__global__ void k_gemm_dual(const float* __restrict__ h,
                            const float* __restrict__ W,
                            const float* __restrict__ resW,
                            const float* __restrict__ resb,
                            float* __restrict__ hout,
                            float* __restrict__ res,
                            int N) {
    const int wave = threadIdx.x >> 5;       // 0..7 -> col tile
    const int lane = threadIdx.x & 31;
    const int half = lane >> 4;              // 0: K={0,1}, 1: K={2,3}
    const int l    = lane & 15;
    const int row0 = blockIdx.x * 16;
    const int col0 = wave * 16;
    const int mat  = blockIdx.y;
    if (row0 >= N) return;

    const float* B    = (mat == 0) ? W : resW;
    const int    rrow = min(row0 + l, N - 1);          // clamp (N % 16 == 0 anyway)
    const float* arow = h + (size_t)rrow * D;

    v8f c = (v8f)0.0f;
#pragma unroll
    for (int kk = 0; kk < D; kk += 4) {
        // A 16x4 layout: lanes 0-15 hold (M=l, K=0/1), lanes 16-31 hold (M=l, K=2/3)
        v2f a, b;
        a.x = arow[kk + 2 * half + 0];
        a.y = arow[kk + 2 * half + 1];
        // B 4x16 layout: row-striped across lanes; half selects K pair
        b.x = B[(size_t)(kk + 2 * half + 0) * D + col0 + l];
        b.y = B[(size_t)(kk + 2 * half + 1) * D + col0 + l];
        c = __builtin_amdgcn_wmma_f32_16x16x4_f32(
                /*neg_a=*/false, a, /*neg_b=*/false, b,
                /*c_mod=*/(short)0, c, /*reuse_a=*/false, /*reuse_b=*/false);
    }

    // C/D layout: VGPR r -> row = row0 + r + 8*half, col = col0 + (lane&15)
#pragma unroll
    for (int r = 0; r < 8; ++r) {
        int row = row0 + r + 8 * half;
        if (row >= N) continue;
        int col = col0 + l;
        float v = c[r];
        if (mat == 0) {
            if (v != v) v = 0.0f;                       // nan -> 0
            hout[(size_t)row * D + col] = v;
        } else {
            res[(size_t)row * D + col] = v + resb[col]; // residual + bias
        }
    }
}

// ---------------------------------------------------------------------------
// Kernel 2: per-edge attention logit (one wave per edge) + segment max.
// ---------------------------------------------------------------------------
__global__ void k_edge_logit(const float* __restrict__ hout,
                             const int* __restrict__ src,
                             const int* __restrict__ dst,
                             float* __restrict__ e,
                             unsigned* __restrict__ segmax,
                             int E) {
    int widx = blockIdx.x * (blockDim.x >> 5) + (threadIdx.x >> 5);
    if (widx >= E) return;
    int lane = threadIdx.x & 31;
    int s = src[widx], d = dst[widx];
    const float4* ps = (const float4*)(hout + (size_t)s * D);
    const float4* pd = (const float4*)(hout + (size_t)d * D);
    float4 a = ps[lane];
    float4 b = pd[lane];
    float part = a.x * b.x + a.y * b.y + a.z * b.z + a.w * b.w;
#pragma unroll
    for (int off = 16; off > 0; off >>= 1)
        part += __shfl_xor(part, off, 32);
    if (lane == 0) {
        float ev = part > 0.0f ? part : NEG_SLOPE * part;
        e[widx] = ev;
        atomicMax(&segmax[d], f2ord(ev));
    }
}

// ---------------------------------------------------------------------------
// Kernel 3: exp(e - segmax[dst]), accumulate denom and degree.
// ---------------------------------------------------------------------------
__global__ void k_edge_exp(const float* __restrict__ e,
                           const int* __restrict__ dst,
                           const unsigned* __restrict__ segmax,
                           float* __restrict__ exb,
                           float* __restrict__ denom,
                           float* __restrict__ deg,
                           int E) {
    int i = blockIdx.x * blockDim.x + threadIdx.x;
    if (i >= E) return;
    int d = dst[i];
    float v = expf(e[i] - ord2f(segmax[d]));
    exb[i] = v;
    atomicAdd(&denom[d], v);
    atomicAdd(&deg[d], 1.0f);
}

// ---------------------------------------------------------------------------
// Kernel 4: weighted scatter-sum of messages (one wave per edge).
// ---------------------------------------------------------------------------
__global__ void k_edge_scatter(const float* __restrict__ hout,
                               const int* __restrict__ src,
                               const int* __restrict__ dst,
                               const float* __restrict__ exb,
                               const float* __restrict__ denom,
                               float* __restrict__ agg,
                               int E) {
    int widx = blockIdx.x * (blockDim.x >> 5) + (threadIdx.x >> 5);
    if (widx >= E) return;
    int lane = threadIdx.x & 31;
    int s = src[widx], d = dst[widx];
    float att = exb[widx] / denom[d];
    const float4* ps = (const float4*)(hout + (size_t)s * D);
    float4 a = ps[lane];
    float* out = agg + (size_t)d * D + lane * 4;
    atomicAdd(out + 0, att * a.x);
    atomicAdd(out + 1, att * a.y);
    atomicAdd(out + 2, att * a.z);
    atomicAdd(out + 3, att * a.w);
}

// ---------------------------------------------------------------------------
// Kernel 5: mean + residual + ELU, in place in d_out.
// ---------------------------------------------------------------------------
__global__ void k_finalize(const float* __restrict__ res,
                           const float* __restrict__ deg,
                           float* __restrict__ out,
                           int total) {
    int i = blockIdx.x * blockDim.x + threadIdx.x;
    if (i >= total) return;
    int n = i >> 7;   // / 128
    float dg   = deg[n];
    float v    = out[i];
    float mean = dg > 0.0f ? v / fmaxf(dg, 1.0f) : 0.0f;
    float r    = mean + res[i];
    out[i] = r > 0.0f ? r : (expf(r) - 1.0f);
}

// ---------------------------------------------------------------------------
extern "C" void kernel_launch(void* const* d_in, const int* in_sizes, int n_in,
                              void* d_out, int out_size, void* d_ws, size_t ws_size,
                              hipStream_t stream) {
    const float* h    = (const float*)d_in[0];
    const float* W    = (const float*)d_in[1];
    const float* resW = (const float*)d_in[2];
    const float* resb = (const float*)d_in[3];
    const int*   src  = (const int*)d_in[4];
    const int*   dst  = (const int*)d_in[5];

    const int N  = in_sizes[0] / D;
    const int E  = in_sizes[4];
    const int ND = N * D;

    // workspace carve-up
    char* ws = (char*)d_ws;
    float*    hout   = (float*)ws;    ws += (size_t)ND * sizeof(float);
    float*    res    = (float*)ws;    ws += (size_t)ND * sizeof(float);
    float*    e      = (float*)ws;    ws += (size_t)E * sizeof(float);
    float*    exb    = (float*)ws;    ws += (size_t)E * sizeof(float);
    unsigned* segmax = (unsigned*)ws; ws += (size_t)N * sizeof(unsigned);
    float*    denom  = (float*)ws;    ws += (size_t)N * sizeof(float);
    float*    deg    = (float*)ws;    ws += (size_t)N * sizeof(float);
    float*    agg    = (float*)d_out;

    // 0: init
    {
        int threads = 256;
        int blocks  = (ND + threads - 1) / threads;
        k_init<<<blocks, threads, 0, stream>>>(segmax, denom, deg, agg, N, ND);
    }
    // 1: dual GEMM (WMMA fp32)
    {
        dim3 grid((N + 15) / 16, 2);
        k_gemm_dual<<<grid, 256, 0, stream>>>(h, W, resW, resb, hout, res, N);
    }
    // 2: edge logits + segment max (wave per edge)
    {
        int wpb    = 8;                    // 256 threads = 8 waves
        int blocks = (E + wpb - 1) / wpb;
        k_edge_logit<<<blocks, 256, 0, stream>>>(hout, src, dst, e, segmax, E);
    }
    // 3: exp + denom/deg
    {
        int threads = 256;
        int blocks  = (E + threads - 1) / threads;
        k_edge_exp<<<blocks, threads, 0, stream>>>(e, dst, segmax, exb, denom, deg, E);
    }
    // 4: weighted scatter (wave per edge)
    {
        int wpb    = 8;
        int blocks = (E + wpb - 1) / wpb;
        k_edge_scatter<<<blocks, 256, 0, stream>>>(hout, src, dst, exb, denom, agg, E);
    }
    // 5: finalize
    {
        int threads = 256;
        int blocks  = (ND + threads - 1) / threads;
        k_finalize<<<blocks, threads, 0, stream>>>(res, deg, agg, ND);
    }
}